// Interpolator_75436805587560
// MI455X (gfx1250) — compile-verified
//
#include <hip/hip_runtime.h>
#include <cstdint>
#include <cstddef>

// Trilinear interpolation: 129^3 x 16 f32 table, N query points in [0,1)^3.
// Layout: 4 lanes per point (lane q in 0..3 owns features 4q..4q+3).
//   - every corner gather = one contiguous 64B transaction (4 lanes x b128)
//   - output = contiguous 64B per point, fully coalesced per wave
// Positions are staged into LDS with CDNA5 async global->LDS loads
// (ASYNCcnt path), then broadcast-read by the 4 lanes of each point.

#define THREADS          256
#define POINTS_PER_BLOCK 64                  // THREADS / 4
#define POS_CHUNKS       48                  // 64 pts * 3 floats * 4B / 16B

#define GRIDF   128.0f
#define STRIDE_Y 129
#define STRIDE_X (129 * 129)                 // 16641

__global__ __launch_bounds__(THREADS)
void Interpolator_75436805587560_kernel(const float* __restrict__ pos,
                                        const float* __restrict__ table,
                                        float* __restrict__ out,
                                        int n_points)
{
    __shared__ float s_pos[POS_CHUNKS * 4];  // 768 B of positions for this block

    const int t = threadIdx.x;

    // ---------- stage this block's positions into LDS (async-to-LDS) ----------
    {
        // total 16B chunks covering all valid position floats (ceil)
        const int total_chunks = (n_points * 3 + 3) >> 2;
        if (t < POS_CHUNKS) {
            int gchunk = blockIdx.x * POS_CHUNKS + t;
            if (gchunk >= total_chunks) gchunk = total_chunks - 1;  // clamp, no OOB
            const uint32_t lds_off = (uint32_t)(uintptr_t)(&s_pos[t * 4]);
            const uint32_t goff    = (uint32_t)gchunk * 16u;
            asm volatile(
                "global_load_async_to_lds_b128 %0, %1, %2"
                :
                : "v"(lds_off), "v"(goff), "s"(pos)
                : "memory");
        }
        asm volatile("s_wait_asynccnt 0" ::: "memory");
        __syncthreads();
    }

    const int lp = t >> 2;                       // local point 0..63
    const int q  = t & 3;                        // feature quad 0..3
    const int pt = blockIdx.x * POINTS_PER_BLOCK + lp;
    if (pt >= n_points) return;

    const float px = s_pos[lp * 3 + 0];
    const float py = s_pos[lp * 3 + 1];
    const float pz = s_pos[lp * 3 + 2];

    // lower corner + fractional coords (bit-identical to reference math:
    // steps = 1/128 is a power of two, so p/steps == p*128 exactly)
    const float tx = px * GRIDF, ty = py * GRIDF, tz = pz * GRIDF;
    const int   lx = (int)floorf(tx), ly = (int)floorf(ty), lz = (int)floorf(tz);
    const float fx = tx - (float)lx, fy = ty - (float)ly, fz = tz - (float)lz;
    const float ax = 1.0f - fx, ay = 1.0f - fy, az = 1.0f - fz;

    // 8 trilinear weights, corner bit order (bx,by,bz) matching BIT_TABLE/strides
    const float wyz00 = ay * az, wyz01 = ay * fz, wyz10 = fy * az, wyz11 = fy * fz;
    const float w0 = ax * wyz00;   // (0,0,0)
    const float w1 = ax * wyz01;   // (0,0,1)
    const float w2 = ax * wyz10;   // (0,1,0)
    const float w3 = ax * wyz11;   // (0,1,1)
    const float w4 = fx * wyz00;   // (1,0,0)
    const float w5 = fx * wyz01;   // (1,0,1)
    const float w6 = fx * wyz10;   // (1,1,0)
    const float w7 = fx * wyz11;   // (1,1,1)

    const int base = lx * STRIDE_X + ly * STRIDE_Y + lz;

    // table rows are 16 floats = 4 float4s; this lane reads quad q of each corner
    const float4* __restrict__ tb = (const float4*)table;
    const size_t b = (size_t)base * 4 + (size_t)q;

    // issue all 8 gathers (each a 64B-coalesced b128 across the 4 lanes of a point)
    const float4 v0 = tb[b];
    const float4 v1 = tb[b + (size_t)1 * 4];                 // +z
    const float4 v2 = tb[b + (size_t)STRIDE_Y * 4];          // +y
    const float4 v3 = tb[b + (size_t)(STRIDE_Y + 1) * 4];    // +y+z
    const float4 v4 = tb[b + (size_t)STRIDE_X * 4];          // +x
    const float4 v5 = tb[b + (size_t)(STRIDE_X + 1) * 4];    // +x+z
    const float4 v6 = tb[b + (size_t)(STRIDE_X + STRIDE_Y) * 4];
    const float4 v7 = tb[b + (size_t)(STRIDE_X + STRIDE_Y + 1) * 4];

    float4 acc;
    acc.x = w0 * v0.x; acc.y = w0 * v0.y; acc.z = w0 * v0.z; acc.w = w0 * v0.w;
    acc.x = fmaf(w1, v1.x, acc.x); acc.y = fmaf(w1, v1.y, acc.y);
    acc.z = fmaf(w1, v1.z, acc.z); acc.w = fmaf(w1, v1.w, acc.w);
    acc.x = fmaf(w2, v2.x, acc.x); acc.y = fmaf(w2, v2.y, acc.y);
    acc.z = fmaf(w2, v2.z, acc.z); acc.w = fmaf(w2, v2.w, acc.w);
    acc.x = fmaf(w3, v3.x, acc.x); acc.y = fmaf(w3, v3.y, acc.y);
    acc.z = fmaf(w3, v3.z, acc.z); acc.w = fmaf(w3, v3.w, acc.w);
    acc.x = fmaf(w4, v4.x, acc.x); acc.y = fmaf(w4, v4.y, acc.y);
    acc.z = fmaf(w4, v4.z, acc.z); acc.w = fmaf(w4, v4.w, acc.w);
    acc.x = fmaf(w5, v5.x, acc.x); acc.y = fmaf(w5, v5.y, acc.y);
    acc.z = fmaf(w5, v5.z, acc.z); acc.w = fmaf(w5, v5.w, acc.w);
    acc.x = fmaf(w6, v6.x, acc.x); acc.y = fmaf(w6, v6.y, acc.y);
    acc.z = fmaf(w6, v6.z, acc.z); acc.w = fmaf(w6, v6.w, acc.w);
    acc.x = fmaf(w7, v7.x, acc.x); acc.y = fmaf(w7, v7.y, acc.y);
    acc.z = fmaf(w7, v7.z, acc.z); acc.w = fmaf(w7, v7.w, acc.w);

    float4* __restrict__ o = (float4*)out;
    o[(size_t)pt * 4 + (size_t)q] = acc;
}

extern "C" void kernel_launch(void* const* d_in, const int* in_sizes, int n_in,
                              void* d_out, int out_size, void* d_ws, size_t ws_size,
                              hipStream_t stream) {
    (void)n_in; (void)out_size; (void)d_ws; (void)ws_size;
    const float* pos   = (const float*)d_in[0];   // [N,3] f32
    const float* table = (const float*)d_in[1];   // [129^3,16] f32
    float*       out   = (float*)d_out;           // [N,16] f32

    const int n_points = in_sizes[0] / 3;
    const int grid     = (n_points + POINTS_PER_BLOCK - 1) / POINTS_PER_BLOCK;

    Interpolator_75436805587560_kernel<<<dim3(grid), dim3(THREADS), 0, stream>>>(
        pos, table, out, n_points);
}